// FCDiscriminator_68934225101047
// MI455X (gfx1250) — compile-verified
//
#include <hip/hip_runtime.h>

// ---------------------------------------------------------------------------
// CDNA5 (gfx1250) fp32 implicit-GEMM 3D UNet.
// Every convolution is a WMMA F32_16x16x4 implicit GEMM:
//   M = 16 output voxels per wave, N = 16 output channels, K = taps * Cin.
// BN+ReLU fused into the A-fragment load. Per-tap weight slices are
// double-buffered in LDS (one barrier per tap, staging overlaps compute).
// All spatial dims are powers of two -> decode via shifts (no div/mod).
// Kernel is template-specialized on <MODE, KSZ, FULLK, HAS_BN>; halo/tail
// masking is branchless (clamped address + post-BN 0/1 mask multiply).
// Deconvolution (2x2x2 stride 2) runs as 8 parity sub-grids (blockIdx.z=tap).
// ---------------------------------------------------------------------------

typedef float v2f __attribute__((ext_vector_type(2)));
typedef float v8f __attribute__((ext_vector_type(8)));

#define MAX_CK 112   // max K-extent per tap (2*56 channels)

struct ConvArgs {
  const float* __restrict__ in;        // activations, [voxel][in_stride] fp32
  const float* __restrict__ w;         // weights (kz,ky,kx,Cin,Cout) row-major
  float*       __restrict__ out;       // [voxel][out_stride]
  const float* __restrict__ bn_scale;  // folded BN scale (if HAS_BN)
  const float* __restrict__ bn_shift;
  int lg;                 // log2 of input spatial dim (D=H=W=1<<lg), N=4
  int olg;                // log2 of output spatial dim
  int Cin_real, Cin_k;    // real input channels; K-extent per tap (mult of 4)
  int in_stride;
  int Cout_real, out_stride, out_off, out_cap;
};

__device__ __forceinline__ v2f load_b(const float* lw, int c, int col) {
  // paired LDS layout: element (c,n) at ((c>>1)<<5) + (n<<1) + (c&1); c even
  return *(const v2f*)(lw + ((c >> 1) << 5) + (col << 1));
}

template <bool FULLK, bool HAS_BN>
__device__ __forceinline__ v2f load_a(const float* inrow, int c, float msk,
                                      int cin_real, const float* ls,
                                      const float* lh) {
  v2f r;
  if (FULLK) {
    r = *(const v2f*)(inrow + c);              // channels c, c+1 contiguous
    if (HAS_BN) {
      const v2f s = *(const v2f*)(ls + c);
      const v2f h = *(const v2f*)(lh + c);
      r.x = fmaxf(fmaf(r.x, s.x, h.x), 0.f);   // BN + ReLU fused
      r.y = fmaxf(fmaf(r.y, s.y, h.y), 0.f);
    }
    r.x *= msk; r.y *= msk;                    // branchless halo/tail mask
  } else {  // stem only (Cin_real=1, no BN): clamped index + channel mask
    const float m0 = (c     < cin_real) ? msk : 0.f;
    const float m1 = (c + 1 < cin_real) ? msk : 0.f;
    const int   c0 = (c     < cin_real) ? c     : 0;
    const int   c1 = (c + 1 < cin_real) ? c + 1 : 0;
    r.x = inrow[c0] * m0;
    r.y = inrow[c1] * m1;
  }
  return r;
}

// MODE: 0 = conv (M enumerates outputs), 1 = 2x2x2 deconv (tap = blockIdx.z)
template <int MODE, int KSZ, bool FULLK, bool HAS_BN>
__global__ __launch_bounds__(128) void conv3d_wmma_f32(ConvArgs a) {
  constexpr int NTAPS  = (MODE == 0) ? KSZ * KSZ * KSZ : 1;
  constexpr int STRIDE = (MODE == 0 && KSZ == 2) ? 2 : 1;
  constexpr int PAD    = (MODE == 0 && KSZ == 3) ? 1 : 0;

  __shared__ float lds_w[2 * MAX_CK * 16];  // double-buffered tap weight slice
  __shared__ float lds_scale[MAX_CK];
  __shared__ float lds_shift[MAX_CK];

  const int lane    = threadIdx.x & 31;
  const int wave    = threadIdx.x >> 5;
  const int hi      = lane >> 4;       // half-wave selector (K offset 0 / 2)
  const int col     = lane & 15;       // M row (A) / N col (B, D)
  const int ntile16 = blockIdx.y * 16;

  if (HAS_BN) {
    for (int c = threadIdx.x; c < a.Cin_real; c += blockDim.x) {
      lds_scale[c] = a.bn_scale[c];
      lds_shift[c] = a.bn_shift[c];
    }
  }

  const int mg     = (MODE == 0) ? a.olg : a.lg;   // grid M enumerates
  const int Dm     = 1 << mg;
  const int Din    = 1 << a.lg;
  const int NVm    = 4 << (3 * mg);
  const int vbase  = (blockIdx.x * 4 + wave) * 16;
  const int midx   = vbase + col;
  const bool mvalid = (midx < NVm);

  // decode lane's M-row voxel (shifts only; dims are powers of two)
  const int mx = midx & (Dm - 1);
  const int my = (midx >> mg) & (Dm - 1);
  const int mz = (midx >> (2 * mg)) & (Dm - 1);
  const int mn = midx >> (3 * mg);

  const int kb     = 2 * hi;
  const int wslice = a.Cin_real * a.Cout_real;

  // cooperative stage of one tap's weight slice into LDS buffer `buf`
  auto stage = [&](int buf, int tap) {
    const int  ne    = a.Cin_k * 16;
    const long wbase = (long)tap * wslice;
    float* dst = lds_w + buf * (MAX_CK * 16);
    for (int e = threadIdx.x; e < ne; e += blockDim.x) {
      const int cc = e >> 4;
      const int n  = e & 15;
      const int co = ntile16 + n;
      float wv = 0.f;
      if (cc < a.Cin_real && co < a.Cout_real)
        wv = a.w[wbase + (long)cc * a.Cout_real + co];
      dst[((cc >> 1) << 5) + (n << 1) + (cc & 1)] = wv;
    }
  };

  const int tap0 = (MODE == 0) ? 0 : (int)blockIdx.z;
  stage(0, tap0);
  __syncthreads();

  v8f acc = {0.f, 0.f, 0.f, 0.f, 0.f, 0.f, 0.f, 0.f};

  int tz = 0, ty = 0, tx = 0;  // incremental tap coords (MODE 0)
  for (int t = 0; t < NTAPS; ++t) {
    // input voxel for this lane's M row under this tap (branchless mask)
    bool ivalid;
    long ivox;
    if (MODE == 0) {
      const int iz = mz * STRIDE + tz - PAD;
      const int iy = my * STRIDE + ty - PAD;
      const int ix = mx * STRIDE + tx - PAD;
      ivalid = mvalid && (unsigned)iz < (unsigned)Din &&
               (unsigned)iy < (unsigned)Din && (unsigned)ix < (unsigned)Din;
      ivox = (((((long)(mn << a.lg) + iz) << a.lg) + iy) << a.lg) + ix;
    } else {
      ivalid = mvalid;
      ivox   = midx;
    }
    const float msk = ivalid ? 1.f : 0.f;
    const long  svox = ivalid ? ivox : 0;     // clamped: always-safe address

    // overlap: stage next tap into the other buffer while computing this one
    if (t + 1 < NTAPS) {
      stage((t + 1) & 1, t + 1);
      __builtin_prefetch(a.w + (long)(t + 2) * wslice, 0, 1);
    }

    const float* inrow = a.in + svox * (long)a.in_stride;
    const float* lw    = lds_w + (t & 1) * (MAX_CK * 16);

    // software-pipelined K loop: issue next fragment's loads before the wmma
    v2f av = load_a<FULLK, HAS_BN>(inrow, kb, msk, a.Cin_real, lds_scale, lds_shift);
    v2f bv = load_b(lw, kb, col);
    for (int c0 = 0; c0 < a.Cin_k; c0 += 4) {
      v2f av2 = {0.f, 0.f}, bv2 = {0.f, 0.f};
      const int cn = c0 + 4;
      if (cn < a.Cin_k) {
        av2 = load_a<FULLK, HAS_BN>(inrow, cn + kb, msk, a.Cin_real,
                                    lds_scale, lds_shift);
        bv2 = load_b(lw, cn + kb, col);
      }
      acc = __builtin_amdgcn_wmma_f32_16x16x4_f32(
          /*neg_a=*/false, av, /*neg_b=*/false, bv,
          /*c_mod=*/(short)0, acc, /*reuse_a=*/false, /*reuse_b=*/false);
      av = av2; bv = bv2;
    }

    __syncthreads();  // next iteration re-stages buffer (t&1)

    if (MODE == 0) {
      if (++tx == KSZ) { tx = 0; if (++ty == KSZ) { ty = 0; ++tz; } }
    }
  }

  // Store D: VGPR v holds M = v + 8*hi, N = col.
  int tz0 = 0, ty0 = 0, tx0 = 0;
  if (MODE == 1) {
    const int t = (int)blockIdx.z;
    tz0 = t >> 2; ty0 = (t >> 1) & 1; tx0 = t & 1;
  }
  const int co = a.out_off + ntile16 + col;
  for (int v = 0; v < 8; ++v) {
    const int row = v + 8 * hi;
    const int om  = vbase + row;
    if (om >= NVm) continue;
    long ovox;
    if (MODE == 0) {
      ovox = om;
    } else {
      const int x = om & (Dm - 1);
      const int y = (om >> mg) & (Dm - 1);
      const int z = (om >> (2 * mg)) & (Dm - 1);
      const int n = om >> (3 * mg);
      ovox = (((((long)(n << a.olg) + (2 * z + tz0)) << a.olg) +
               (2 * y + ty0)) << a.olg) + (2 * x + tx0);
    }
    if (co < a.out_cap) a.out[ovox * (long)a.out_stride + co] = acc[v];
  }
}

// Fold BN params into scale/shift: y = max(x*scale + shift, 0)
__global__ void bn_prep(const float* g, const float* b, const float* m,
                        const float* v, float* scale, float* shift, int C) {
  const int i = threadIdx.x;
  if (i < C) {
    const float s = g[i] * rsqrtf(v[i] + 1e-4f);
    scale[i] = s;
    shift[i] = fmaf(-m[i], s, b[i]);
  }
}

// Final standalone BN+ReLU into the (unpadded) output tensor.
__global__ void bn_relu_out(const float* __restrict__ in, int in_stride,
                            const float* __restrict__ scale,
                            const float* __restrict__ shift,
                            float* __restrict__ out, int C, int NV) {
  const int i = blockIdx.x * blockDim.x + threadIdx.x;
  if (i >= NV * C) return;
  const int c  = i & (C - 1);   // C = 8
  const int vx = i / C;
  const float x = in[(long)vx * in_stride + c];
  out[i] = fmaxf(fmaf(x, scale[c], shift[c]), 0.f);
}

// ---------------------------------------------------------------------------
// Host orchestration
// ---------------------------------------------------------------------------

namespace {

struct BNp { const float *g, *b, *m, *v; };
struct Lvl { BNp b0, b1, dn, up; const float *w0, *w1, *wd, *wu; };

int parse_lvl(void* const* din, int i, Lvl* L, int lvl) {
  // sorted-key pytree order: block0_bn{beta,gamma,mean,var}, block0_w,
  // block1_bn, block1_w, down_bn, down_w, inner..., up_bn, up_w
  auto bn = [&](BNp& p) {
    p.b = (const float*)din[i++]; p.g = (const float*)din[i++];
    p.m = (const float*)din[i++]; p.v = (const float*)din[i++];
  };
  bn(L[lvl].b0); L[lvl].w0 = (const float*)din[i++];
  if (lvl < 6) {
    bn(L[lvl].b1); L[lvl].w1 = (const float*)din[i++];
    bn(L[lvl].dn); L[lvl].wd = (const float*)din[i++];
    i = parse_lvl(din, i, L, lvl + 1);
    bn(L[lvl].up); L[lvl].wu = (const float*)din[i++];
  }
  return i;
}

inline int pad16(int c) { return (c + 15) & ~15; }

struct RunCtx {
  Lvl L[7];
  float* inb[7];  float* catb[7];
  int inS[7], catS[7], pl[7];
  float *b0s[7], *b0h[7], *b1s[7], *b1h[7], *dns[7], *dnh[7], *ups[7], *uph[7];
  hipStream_t stream;
};

enum ConvKind { K_STEM, K_CONV3, K_DOWN, K_UP };

void launch_conv(const ConvArgs& a, int ntiles, ConvKind kind, int lg_m,
                 hipStream_t stream) {
  const int NVm = 4 << (3 * lg_m);
  dim3 grid((NVm + 63) / 64, ntiles, kind == K_UP ? 8 : 1);
  switch (kind) {
    case K_STEM:  conv3d_wmma_f32<0, 3, false, false><<<grid, 128, 0, stream>>>(a); break;
    case K_CONV3: conv3d_wmma_f32<0, 3, true,  true ><<<grid, 128, 0, stream>>>(a); break;
    case K_DOWN:  conv3d_wmma_f32<0, 2, true,  true ><<<grid, 128, 0, stream>>>(a); break;
    case K_UP:    conv3d_wmma_f32<1, 2, true,  true ><<<grid, 128, 0, stream>>>(a); break;
  }
}

void run_level(RunCtx& c, int lvl) {
  const int p = c.pl[lvl], lg = 6 - lvl;

  // block0: bn_relu(in) -> 3x3x3 conv -> catb[lvl] channels [0,p)
  ConvArgs a{};
  a.in = c.inb[lvl]; a.w = c.L[lvl].w0; a.out = c.catb[lvl];
  a.bn_scale = c.b0s[lvl]; a.bn_shift = c.b0h[lvl];
  a.lg = lg; a.olg = lg;
  a.Cin_real = p; a.Cin_k = p; a.in_stride = c.inS[lvl];
  a.Cout_real = p; a.out_stride = c.catS[lvl]; a.out_off = 0; a.out_cap = c.catS[lvl];
  launch_conv(a, (p + 15) / 16, K_CONV3, lg, c.stream);
  if (lvl == 6) return;  // deepest level: result = catb[6]

  const int p1 = c.pl[lvl + 1];

  // down: bn_relu(block0) -> 2x2x2 stride-2 conv -> inb[lvl+1]
  a = ConvArgs{};
  a.in = c.catb[lvl]; a.w = c.L[lvl].wd; a.out = c.inb[lvl + 1];
  a.bn_scale = c.dns[lvl]; a.bn_shift = c.dnh[lvl];
  a.lg = lg; a.olg = lg - 1;
  a.Cin_real = p; a.Cin_k = p; a.in_stride = c.catS[lvl];
  a.Cout_real = p1; a.out_stride = c.inS[lvl + 1]; a.out_off = 0;
  a.out_cap = c.inS[lvl + 1];
  launch_conv(a, (p1 + 15) / 16, K_DOWN, lg - 1, c.stream);

  run_level(c, lvl + 1);

  const float* res = (lvl + 1 == 6) ? c.catb[6] : c.inb[lvl + 1];
  const int resS   = (lvl + 1 == 6) ? c.catS[6] : c.inS[lvl + 1];

  // up: bn_relu(inner) -> 2x2x2 deconv -> catb[lvl] channels [p,2p)
  a = ConvArgs{};
  a.in = res; a.w = c.L[lvl].wu; a.out = c.catb[lvl];
  a.bn_scale = c.ups[lvl]; a.bn_shift = c.uph[lvl];
  a.lg = lg - 1; a.olg = lg;
  a.Cin_real = p1; a.Cin_k = (p1 + 3) & ~3; a.in_stride = resS;
  a.Cout_real = p; a.out_stride = c.catS[lvl]; a.out_off = p; a.out_cap = c.catS[lvl];
  launch_conv(a, (p + 15) / 16, K_UP, lg - 1, c.stream);

  // block1: bn_relu(concat 2p) -> 3x3x3 conv -> inb[lvl] (level result)
  a = ConvArgs{};
  a.in = c.catb[lvl]; a.w = c.L[lvl].w1; a.out = c.inb[lvl];
  a.bn_scale = c.b1s[lvl]; a.bn_shift = c.b1h[lvl];
  a.lg = lg; a.olg = lg;
  a.Cin_real = 2 * p; a.Cin_k = 2 * p; a.in_stride = c.catS[lvl];
  a.Cout_real = p; a.out_stride = c.inS[lvl]; a.out_off = 0; a.out_cap = c.inS[lvl];
  launch_conv(a, (p + 15) / 16, K_CONV3, lg, c.stream);
}

}  // namespace

extern "C" void kernel_launch(void* const* d_in, const int* in_sizes, int n_in,
                              void* d_out, int out_size, void* d_ws, size_t ws_size,
                              hipStream_t stream) {
  (void)in_sizes; (void)n_in; (void)out_size;

  // ---- parse flattened pytree: params{final_bn, stem_w, unet}, x ----
  BNp fin;
  fin.b = (const float*)d_in[0]; fin.g = (const float*)d_in[1];
  fin.m = (const float*)d_in[2]; fin.v = (const float*)d_in[3];
  const float* stem_w = (const float*)d_in[4];

  RunCtx c{};
  c.stream = stream;
  const int idx_after = parse_lvl(d_in, 5, c.L, 0);
  const float* x = (const float*)d_in[idx_after];

  // ---- workspace carve ----
  char* cur = (char*)d_ws;
  auto carve = [&](size_t bytes) -> void* {
    void* p = cur;
    cur += (bytes + 255) & ~(size_t)255;
    return p;
  };

  const int pl_[7] = {8, 16, 24, 32, 40, 48, 56};
  int Dd[7];
  for (int l = 0; l < 7; ++l) {
    c.pl[l]   = pl_[l];
    Dd[l]     = 64 >> l;
    c.inS[l]  = pad16(pl_[l]);
    c.catS[l] = (l < 6) ? 2 * pl_[l] : pad16(pl_[l]);
  }
  size_t need = 0;
  for (int l = 0; l < 7; ++l) {
    const long NV = 4L * Dd[l] * Dd[l] * Dd[l];
    need += (((size_t)NV * c.inS[l] * 4 + 255) & ~(size_t)255);
    need += (((size_t)NV * c.catS[l] * 4 + 255) & ~(size_t)255);
  }
  need += 27 * 2 * 512;  // BN scale/shift tables
  if (ws_size < need) return;  // workspace too small; nothing safe to do

  for (int l = 0; l < 7; ++l) {
    const long NV = 4L * Dd[l] * Dd[l] * Dd[l];
    c.inb[l]  = (float*)carve((size_t)NV * c.inS[l] * 4);
    c.catb[l] = (float*)carve((size_t)NV * c.catS[l] * 4);
  }

  auto prep = [&](const BNp& p, int C, float*& s, float*& h) {
    s = (float*)carve(128 * 4);
    h = (float*)carve(128 * 4);
    bn_prep<<<1, 128, 0, stream>>>(p.g, p.b, p.m, p.v, s, h, C);
  };
  for (int l = 0; l < 7; ++l) {
    prep(c.L[l].b0, c.pl[l], c.b0s[l], c.b0h[l]);
    if (l < 6) {
      prep(c.L[l].b1, 2 * c.pl[l], c.b1s[l], c.b1h[l]);
      prep(c.L[l].dn, c.pl[l],     c.dns[l], c.dnh[l]);
      prep(c.L[l].up, c.pl[l + 1], c.ups[l], c.uph[l]);
    }
  }
  float *fs, *fh;
  prep(fin, 8, fs, fh);

  // ---- stem: 3x3x3 conv 1->8 (no BN on raw input) ----
  {
    ConvArgs a{};
    a.in = x; a.w = stem_w; a.out = c.inb[0];
    a.bn_scale = nullptr; a.bn_shift = nullptr;
    a.lg = 6; a.olg = 6;
    a.Cin_real = 1; a.Cin_k = 4; a.in_stride = 1;
    a.Cout_real = 8; a.out_stride = 16; a.out_off = 0; a.out_cap = 16;
    launch_conv(a, 1, K_STEM, 6, stream);
  }

  // ---- UNet ----
  run_level(c, 0);

  // ---- final BN+ReLU into d_out (4,64,64,64,8) ----
  const int NV0 = 4 * 64 * 64 * 64;
  bn_relu_out<<<(NV0 * 8 + 255) / 256, 256, 0, stream>>>(
      c.inb[0], 16, fs, fh, (float*)d_out, 8, NV0);
}